// RGCNLayer_39840116638008
// MI455X (gfx1250) — compile-verified
//
#include <hip/hip_runtime.h>

#define N_NODES  100000
#define N_EDGESC 100000
#define N_REL    8
#define N_BASES  4
#define IN_FEAT  128
#define OUT_FEAT 128

typedef __attribute__((ext_vector_type(2))) float v2f;
typedef __attribute__((ext_vector_type(8))) float v8f;

// ---------------------------------------------------------------------------
// Wrel[r][i][o] with the reference's scrambled flatten order:
// flat = r*128+i ; i_orig = flat>>3 ; r_orig = flat&7
// Wrel[r][i][o] = sum_b w_comp[r_orig][b] * bases[b][i_orig][o]
// ---------------------------------------------------------------------------
__global__ void build_weights_kernel(const float* __restrict__ bases,
                                     const float* __restrict__ w_comp,
                                     float* __restrict__ Wrel) {
    int t = blockIdx.x * blockDim.x + threadIdx.x;   // [0, 8*128*128)
    if (t >= N_REL * IN_FEAT * OUT_FEAT) return;
    int o    = t & (OUT_FEAT - 1);
    int flat = t >> 7;            // r*128 + i
    int r_o  = flat & 7;
    int i_o  = flat >> 3;
    float s = 0.f;
#pragma unroll
    for (int b = 0; b < N_BASES; ++b)
        s += w_comp[r_o * N_BASES + b] * bases[(b * IN_FEAT + i_o) * OUT_FEAT + o];
    Wrel[t] = s;
}

// ---------------------------------------------------------------------------
// Zero the per-relation aggregation buffer (graph-capture-safe, no memset).
// ---------------------------------------------------------------------------
__global__ void zero_f4_kernel(float4* __restrict__ p, int n4) {
    int t = blockIdx.x * blockDim.x + threadIdx.x;
    if (t < n4) p[t] = make_float4(0.f, 0.f, 0.f, 0.f);
}

// ---------------------------------------------------------------------------
// One wave per edge: coalesced 512B gather of h[src], scale by norm,
// scatter-add into agg[dst] with native f32 atomics (L2-resident).
// ---------------------------------------------------------------------------
__global__ void scatter_edges_kernel(const float* __restrict__ h,
                                     const int*   __restrict__ src,
                                     const int*   __restrict__ dst,
                                     const float* __restrict__ norm,
                                     float* __restrict__ agg) {
    int wave = (blockIdx.x * blockDim.x + threadIdx.x) >> 5;
    int lane = threadIdx.x & 31;
    if (wave >= N_EDGESC) return;
    int   s = src[wave];
    int   d = dst[wave];
    float w = norm[wave];
    const float4 hv = reinterpret_cast<const float4*>(h + (size_t)s * IN_FEAT)[lane];
    float* ap = agg + (size_t)d * IN_FEAT + lane * 4;
    unsafeAtomicAdd(ap + 0, hv.x * w);
    unsafeAtomicAdd(ap + 1, hv.y * w);
    unsafeAtomicAdd(ap + 2, hv.z * w);
    unsafeAtomicAdd(ap + 3, hv.w * w);
}

// ---------------------------------------------------------------------------
// out(+)= agg @ Wr using V_WMMA_F32_16X16X4_F32.
//
// B staging: the full 128x128 Wr (64 KB) is cooperatively loaded once per
// 256-thread workgroup into LDS, PRE-SWIZZLED into WMMA B-fragment order:
//   Bs[(kk*8 + n)*32 + lane] = { Wr[4kk+lh][n*16+ll], Wr[4kk+2+lh][n*16+ll] }
// (lh = lane>>4, ll = lane&15). Inner loop then needs exactly one
// bank-conflict-free ds_load_b64 per (k-step, n-tile): 32 lanes x 2 dwords
// = 64 LDS banks. VMEM path is left free to stream A (global_load_b64).
//
// Wave strip: M = wave*16 .. +16, N = 0..128 (8 tiles of 16).
// A frag (16x4 f32): lane<16 -> row M=lane, cols {k0,k0+1};
//                    lane>=16 -> row M=lane-16, cols {k0+2,k0+3}.
// C/D (16x16 f32): VGPR v <-> M = m0 + v + 8*lh, lane ll <-> N col.
// first: zero-init C; else accumulate onto existing out. last: fuse ReLU.
// ---------------------------------------------------------------------------
__global__ void __launch_bounds__(256)
rgcn_gemm_kernel(const float* __restrict__ agg,
                 const float* __restrict__ Wr,
                 float* __restrict__ out,
                 int first, int last) {
    __shared__ float2 Bs[32 * 8 * 32];   // [kk][n][lane], 64 KB

    for (int e = threadIdx.x; e < 32 * 8 * 32; e += 256) {
        int lane = e & 31;
        int n    = (e >> 5) & 7;
        int kk   = e >> 8;
        int lh = lane >> 4, ll = lane & 15;
        int col = n * 16 + ll;
        float2 v;
        v.x = Wr[(4 * kk + lh) * OUT_FEAT + col];
        v.y = Wr[(4 * kk + 2 + lh) * OUT_FEAT + col];
        Bs[e] = v;
    }
    __syncthreads();

    int wave = (blockIdx.x * blockDim.x + threadIdx.x) >> 5;
    int lane = threadIdx.x & 31;
    if (wave >= N_NODES / 16) return;          // wave-uniform: EXEC all-1s below
    int m0 = wave * 16;
    int lh = lane >> 4;                        // lane half (0/1)
    int ll = lane & 15;

    v8f acc[8] = {};
    if (!first) {
#pragma unroll
        for (int n = 0; n < 8; ++n)
#pragma unroll
            for (int v = 0; v < 8; ++v)
                acc[n][v] = out[(size_t)(m0 + 8 * lh + v) * OUT_FEAT + n * 16 + ll];
    }

    const float* arow = agg + (size_t)(m0 + ll) * IN_FEAT + 2 * lh;
#pragma unroll 4
    for (int kk = 0; kk < IN_FEAT / 4; ++kk) {
        float2 av = *reinterpret_cast<const float2*>(arow + 4 * kk);
        v2f a; a.x = av.x; a.y = av.y;
#pragma unroll
        for (int n = 0; n < 8; ++n) {
            float2 bv = Bs[(kk * 8 + n) * 32 + lane];
            v2f b; b.x = bv.x; b.y = bv.y;
            acc[n] = __builtin_amdgcn_wmma_f32_16x16x4_f32(
                false, a, false, b, (short)0, acc[n], false, false);
        }
    }

#pragma unroll
    for (int n = 0; n < 8; ++n)
#pragma unroll
        for (int v = 0; v < 8; ++v) {
            float x = acc[n][v];
            if (last) x = fmaxf(x, 0.f);
            out[(size_t)(m0 + 8 * lh + v) * OUT_FEAT + n * 16 + ll] = x;
        }
}

// ---------------------------------------------------------------------------
extern "C" void kernel_launch(void* const* d_in, const int* in_sizes, int n_in,
                              void* d_out, int out_size, void* d_ws, size_t ws_size,
                              hipStream_t stream) {
    const float* h         = (const float*)d_in[0];
    const int*   edge_src  = (const int*)  d_in[1];
    const int*   edge_dst  = (const int*)  d_in[2];
    const float* edge_norm = (const float*)d_in[3];
    const float* bases     = (const float*)d_in[4];
    const float* w_comp    = (const float*)d_in[5];
    float* out = (float*)d_out;

    float* Wrel = (float*)d_ws;                                  // 8*128*128 f32
    float* agg  = Wrel + (size_t)N_REL * IN_FEAT * OUT_FEAT;     // 100000*128 f32
    size_t need = ((size_t)N_REL * IN_FEAT * OUT_FEAT +
                   (size_t)N_NODES * IN_FEAT) * sizeof(float);
    if (ws_size < need) return;

    build_weights_kernel<<<(N_REL * IN_FEAT * OUT_FEAT + 255) / 256, 256, 0, stream>>>(
        bases, w_comp, Wrel);

    const int n4          = N_NODES * IN_FEAT / 4;
    const int zero_blocks = (n4 + 255) / 256;
    const int scat_blocks = (N_EDGESC * 32 + 255) / 256;
    const int gemm_waves  = N_NODES / 16;                         // 6250
    const int gemm_blocks = (gemm_waves + 7) / 8;                 // 8 waves/block

    for (int r = 0; r < N_REL; ++r) {
        zero_f4_kernel<<<zero_blocks, 256, 0, stream>>>((float4*)agg, n4);
        scatter_edges_kernel<<<scat_blocks, 256, 0, stream>>>(
            h,
            edge_src  + (size_t)r * N_EDGESC,
            edge_dst  + (size_t)r * N_EDGESC,
            edge_norm + (size_t)r * N_EDGESC,
            agg);
        rgcn_gemm_kernel<<<gemm_blocks, 256, 0, stream>>>(
            agg, Wrel + (size_t)r * IN_FEAT * OUT_FEAT, out,
            (r == 0) ? 1 : 0, (r == N_REL - 1) ? 1 : 0);
    }
}